// ChamferLoss_86887188398388
// MI455X (gfx1250) — compile-verified
//
#include <hip/hip_runtime.h>
#include <hip/hip_bf16.h>

typedef float v2f __attribute__((ext_vector_type(2)));
typedef float v8f __attribute__((ext_vector_type(8)));

#define TILE 16
#define CHUNKS 8
#define UNROLL 8
#define WAVES_PER_BLOCK 8
#define BLOCK (WAVES_PER_BLOCK * 32)

// Single-instruction min (values are finite; avoids fminf's canonicalize maxes).
__device__ __forceinline__ float fmin_fast(float a, float b) {
    float r;
    asm("v_min_num_f32 %0, %1, %2" : "=v"(r) : "v"(a), "v"(b));
    return r;
}

// Seed running-min buffers with +inf (as uint bits).
__global__ void chamfer_init_kernel(unsigned int* __restrict__ buf, int n) {
    int i = blockIdx.x * blockDim.x + threadIdx.x;
    if (i < n) buf[i] = 0x7F800000u; // +inf
}

// Precompute packed B-side operand per point: [-2x, -2y, -2z, |p|^2].
__global__ void chamfer_pack_kernel(const float* __restrict__ P,
                                    float4* __restrict__ pk, int n) {
    int i = blockIdx.x * blockDim.x + threadIdx.x;
    if (i < n) {
        const float x = P[i * 3 + 0];
        const float y = P[i * 3 + 1];
        const float z = P[i * 3 + 2];
        pk[i] = make_float4(-2.0f * x, -2.0f * y, -2.0f * z, x * x + y * y + z * z);
    }
}

// For each query row q in Q[nQ,3]: min over all r (packed in Rpk[nR]) of
// ||q-r||^2, atomically min'd (uint bit order, clamped >= 0) into rowmin[q].
//
// V_WMMA_F32_16X16X4_F32 with bias folding:
//   A = [qx qy qz 1] (16x4), B = [-2rx -2ry -2rz r2] (4x16), C = q2 per row
//   => D[m][n] = |q_m|^2 + |r_n|^2 - 2 q_m . r_n
__global__ __launch_bounds__(BLOCK) void chamfer_min_kernel(
    const float* __restrict__ Q, const float4* __restrict__ Rpk,
    unsigned int* __restrict__ rowmin, int nQ, int nR)
{
    // Wave-uniform values pinned to SGPRs so the sweep loop gets scalar control.
    const int waveInBlk = __builtin_amdgcn_readfirstlane(threadIdx.x >> 5);
    const int wave      = blockIdx.x * WAVES_PER_BLOCK + waveInBlk;
    const int lane      = threadIdx.x & 31;
    const int lane16    = lane & 15;
    const int half      = lane >> 4;

    const int tilesQ  = nQ / TILE;
    const int tilesR  = nR / TILE;
    const int rowTile = wave / CHUNKS;
    const int chunk   = wave % CHUNKS;
    if (rowTile >= tilesQ) return; // scalar branch; grid sized exactly anyway

    const int tpc   = tilesR / CHUNKS;          // tiles per chunk (even split)
    const int t0    = chunk * tpc;
    const int count = (chunk == CHUNKS - 1) ? (tilesR - t0) : tpc; // last takes rest
    const int mainCount = count & ~(UNROLL - 1);

    const int rowBase = rowTile * TILE;

    // ---- A matrix (16x4 f32, 2 VGPRs/lane), loop-invariant ----
    // lanes 0-15:  M=lane,    VGPR0=K0(qx), VGPR1=K1(qy)
    // lanes 16-31: M=lane-16, VGPR0=K2(qz), VGPR1=K3(1.0)
    const int qr = rowBase + lane16;
    const float qx = Q[qr * 3 + 0];
    const float qy = Q[qr * 3 + 1];
    const float qz = Q[qr * 3 + 2];
    v2f a;
    a.x = half ? qz   : qx;
    a.y = half ? 1.0f : qy;

    // ---- C matrix: c[v] = |q|^2 for row (rowBase + v + half*8), loop-invariant ----
    v8f cmat;
#pragma unroll
    for (int v = 0; v < 8; ++v) {
        const int r = rowBase + v + half * 8;
        const float x = Q[r * 3 + 0];
        const float y = Q[r * 3 + 1];
        const float z = Q[r * 3 + 2];
        cmat[v] = x * x + y * y + z * z;
    }

    v8f rmin;
#pragma unroll
    for (int v = 0; v < 8; ++v) rmin[v] = __builtin_inff();

    // ---- Sweep target tiles ----
    // Packed layout: Rpk[j] = {-2x,-2y,-2z,r2}. Per lane: the float2 at
    // element (j*2 + half):  half=0 -> (K0,K1) = (-2x,-2y)
    //                        half=1 -> (K2,K3) = (-2z, r2)
    // Per tile stride in v2f elements: 2*TILE (= 256 bytes, fits imm offset).
    const v2f* __restrict__ Bp = (const v2f*)Rpk + ((t0 * TILE + lane16) * 2 + half);

    // Main loop: batch UNROLL independent b64 loads, then UNROLL WMMA+min groups.
    for (int tt = 0; tt < mainCount; tt += UNROLL) {
        v2f bb[UNROLL];
#pragma unroll
        for (int k = 0; k < UNROLL; ++k)
            bb[k] = Bp[(tt + k) * (2 * TILE)];
#pragma unroll
        for (int k = 0; k < UNROLL; ++k) {
            v8f d = __builtin_amdgcn_wmma_f32_16x16x4_f32(
                false, a, false, bb[k], (short)0, cmat, false, false);
#pragma unroll
            for (int v = 0; v < 8; ++v) rmin[v] = fmin_fast(rmin[v], d[v]);
        }
    }
    // Scalar tail (empty for 16384-point clouds).
    for (int tt = mainCount; tt < count; ++tt) {
        const v2f b = Bp[tt * (2 * TILE)];
        v8f d = __builtin_amdgcn_wmma_f32_16x16x4_f32(
            false, a, false, b, (short)0, cmat, false, false);
#pragma unroll
        for (int v = 0; v < 8; ++v) rmin[v] = fmin_fast(rmin[v], d[v]);
    }

    // ---- Reduce across the 16 lanes that share each row (xor stays in half) ----
#pragma unroll
    for (int v = 0; v < 8; ++v) {
        float m = rmin[v];
#pragma unroll
        for (int mask = 1; mask < 16; mask <<= 1)
            m = fmin_fast(m, __shfl_xor(m, mask, 32));
        rmin[v] = m;
    }

    if (lane16 == 0) {
#pragma unroll
        for (int v = 0; v < 8; ++v) {
            const int r = rowBase + v + half * 8;
            const float m = fmaxf(rmin[v], 0.0f); // clamp like reference max(sq,0)
            atomicMin(&rowmin[r], __float_as_uint(m)); // nonneg float bits: uint order
        }
    }
}

// Single-block epilogue: mean(sqrt(min)) for both directions, average them.
__global__ void chamfer_finalize_kernel(const unsigned int* __restrict__ fwd,
                                        const unsigned int* __restrict__ bwd,
                                        float* __restrict__ out, int n, int m) {
    __shared__ float sF[256];
    __shared__ float sB[256];
    float aF = 0.0f, aB = 0.0f;
    for (int i = threadIdx.x; i < n; i += 256) aF += sqrtf(__uint_as_float(fwd[i]));
    for (int i = threadIdx.x; i < m; i += 256) aB += sqrtf(__uint_as_float(bwd[i]));
    sF[threadIdx.x] = aF;
    sB[threadIdx.x] = aB;
    __syncthreads();
    for (int s = 128; s > 0; s >>= 1) {
        if (threadIdx.x < s) {
            sF[threadIdx.x] += sF[threadIdx.x + s];
            sB[threadIdx.x] += sB[threadIdx.x + s];
        }
        __syncthreads();
    }
    if (threadIdx.x == 0)
        out[0] = 0.5f * (sF[0] / (float)n + sB[0] / (float)m);
}

extern "C" void kernel_launch(void* const* d_in, const int* in_sizes, int n_in,
                              void* d_out, int out_size, void* d_ws, size_t ws_size,
                              hipStream_t stream) {
    const float* pred   = (const float*)d_in[0];
    const float* target = (const float*)d_in[1];
    const int N = in_sizes[0] / 3; // 16384
    const int M = in_sizes[1] / 3; // 16384

    // Workspace layout:
    //   [ float4 predPk[N] | float4 tgtPk[M] | uint fwd[N] | uint bwd[M] ]
    float4* predPk = (float4*)d_ws;
    float4* tgtPk  = predPk + N;
    unsigned int* fwd = (unsigned int*)(tgtPk + M);
    unsigned int* bwd = fwd + N;

    chamfer_pack_kernel<<<(N + 255) / 256, 256, 0, stream>>>(pred, predPk, N);
    chamfer_pack_kernel<<<(M + 255) / 256, 256, 0, stream>>>(target, tgtPk, M);

    const int tot = N + M;
    chamfer_init_kernel<<<(tot + 255) / 256, 256, 0, stream>>>(fwd, tot);

    // Forward: queries = pred rows, refs = packed target
    {
        const int waves  = (N / TILE) * CHUNKS;
        const int blocks = (waves + WAVES_PER_BLOCK - 1) / WAVES_PER_BLOCK;
        chamfer_min_kernel<<<blocks, BLOCK, 0, stream>>>(pred, tgtPk, fwd, N, M);
    }
    // Backward: queries = target rows, refs = packed pred
    {
        const int waves  = (M / TILE) * CHUNKS;
        const int blocks = (waves + WAVES_PER_BLOCK - 1) / WAVES_PER_BLOCK;
        chamfer_min_kernel<<<blocks, BLOCK, 0, stream>>>(target, predPk, bwd, M, N);
    }

    chamfer_finalize_kernel<<<1, 256, 0, stream>>>(fwd, bwd, (float*)d_out, N, M);
}